// GCNLayer_89103391522827
// MI455X (gfx1250) — compile-verified
//
#include <hip/hip_runtime.h>
#include <hip/hip_bf16.h>

typedef __attribute__((ext_vector_type(2))) float v2f;
typedef __attribute__((ext_vector_type(8))) float v8f;

#define D_IN  256
#define D_OUT 256

// ---------------------------------------------------------------------------
// 1) degree init: deg[i] = 1 (self loop)
// ---------------------------------------------------------------------------
__global__ void deg_init_kernel(float* __restrict__ deg, int N) {
    int i = blockIdx.x * blockDim.x + threadIdx.x;
    if (i < N) deg[i] = 1.0f;
}

// ---------------------------------------------------------------------------
// 2) degree count over dst (edge_index row 1), fp32 atomics into L2
// ---------------------------------------------------------------------------
__global__ void deg_count_kernel(const long long* __restrict__ ei, int E,
                                 float* __restrict__ deg) {
    int e = blockIdx.x * blockDim.x + threadIdx.x;
    if (e < E) {
        long long d = ei[(long long)E + e];  // dst row
        atomicAdd(&deg[d], 1.0f);
    }
}

// ---------------------------------------------------------------------------
// 3) dinv[i] = rsqrt(deg[i])   (deg >= 1 always, so no zero guard needed)
// ---------------------------------------------------------------------------
__global__ void dinv_kernel(float* __restrict__ deg, int N) {
    int i = blockIdx.x * blockDim.x + threadIdx.x;
    if (i < N) deg[i] = __frsqrt_rn(deg[i]);
}

// ---------------------------------------------------------------------------
// 4) h = x @ W   via V_WMMA_F32_16X16X4_F32 (fp32, matches reference numerics)
//    256 threads = 8 waves; wave w computes rows [blk*128 + w*16, +16) x 64 cols.
//    Per K-step of 4: one A frag (float2/lane) reused across 4 B tiles.
//    Fragment layouts per CDNA5 ISA 7.12.2:
//      A 16x4:  lanes 0-15 hold K={0,1}, lanes 16-31 hold K={2,3}; M = lane&15
//      B 4x16:  VGPR0 = rows K=0(l0-15)/K=2(l16-31), VGPR1 = K=1/K=3; N = lane&15
//      C 16x16: VGPR r -> M = r + 8*(lane>>4), N = lane&15
// ---------------------------------------------------------------------------
__global__ __launch_bounds__(256) void gemm_wmma_kernel(
    const float* __restrict__ x, const float* __restrict__ W,
    float* __restrict__ h, int N) {
    const int wave = threadIdx.x >> 5;
    const int lane = threadIdx.x & 31;
    const int hl   = lane >> 4;       // lane half: 0 or 1
    const int l16  = lane & 15;

    const int row0 = (blockIdx.x * 8 + wave) * 16;
    if (row0 >= N) return;
    const int col0 = blockIdx.y * 64;

    // clamp read row (only matters for a ragged tail; 100000 % 16 == 0)
    const int rrow = (row0 + l16 < N) ? (row0 + l16) : (N - 1);
    const float* __restrict__ xrow = x + (size_t)rrow * D_IN;

    v8f acc0 = {}, acc1 = {}, acc2 = {}, acc3 = {};

    for (int k0 = 0; k0 < D_IN; k0 += 4) {
        const int ka = k0 + 2 * hl;                 // this lane's K pair
        v2f a;
        a.x = xrow[ka + 0];
        a.y = xrow[ka + 1];

        const float* __restrict__ w0 = W + (size_t)(ka + 0) * D_OUT + col0 + l16;
        const float* __restrict__ w1 = W + (size_t)(ka + 1) * D_OUT + col0 + l16;
        v2f b0, b1, b2, b3;
        b0.x = w0[0];  b0.y = w1[0];
        b1.x = w0[16]; b1.y = w1[16];
        b2.x = w0[32]; b2.y = w1[32];
        b3.x = w0[48]; b3.y = w1[48];

        acc0 = __builtin_amdgcn_wmma_f32_16x16x4_f32(false, a, false, b0, (short)0, acc0, false, false);
        acc1 = __builtin_amdgcn_wmma_f32_16x16x4_f32(false, a, false, b1, (short)0, acc1, false, false);
        acc2 = __builtin_amdgcn_wmma_f32_16x16x4_f32(false, a, false, b2, (short)0, acc2, false, false);
        acc3 = __builtin_amdgcn_wmma_f32_16x16x4_f32(false, a, false, b3, (short)0, acc3, false, false);
    }

    const bool full = (row0 + 16 <= N);
#pragma unroll
    for (int r = 0; r < 8; ++r) {
        const int row = row0 + r + 8 * hl;
        if (full || row < N) {
            size_t o = (size_t)row * D_OUT + col0 + l16;
            h[o]      = acc0[r];
            h[o + 16] = acc1[r];
            h[o + 32] = acc2[r];
            h[o + 48] = acc3[r];
        }
    }
}

// ---------------------------------------------------------------------------
// 5) out init = self-loop contribution: out[i,f] = dinv[i]^2 * h[i,f]
// ---------------------------------------------------------------------------
__global__ void selfloop_kernel(const float* __restrict__ h,
                                const float* __restrict__ dinv,
                                float* __restrict__ out, long long total) {
    long long t = (long long)blockIdx.x * blockDim.x + threadIdx.x;
    if (t < total) {
        long long node = t >> 8;  // / 256
        float s = dinv[node];
        out[t] = s * s * h[t];
    }
}

// ---------------------------------------------------------------------------
// 6) edge scatter: one wave32 per edge; 32 lanes x 2 float4 = 256 features.
//    Gather of h[src] is a coalesced 1KB burst; adds are global_atomic_add_f32
//    resolved in L2 (h + out ~ 205 MB vs 192 MB global L2).
// ---------------------------------------------------------------------------
__global__ __launch_bounds__(256) void scatter_kernel(
    const long long* __restrict__ ei, int E,
    const float* __restrict__ h, const float* __restrict__ dinv,
    float* __restrict__ out) {
    const int wave = threadIdx.x >> 5;
    const int lane = threadIdx.x & 31;
    const long long e = (long long)blockIdx.x * 8 + wave;
    if (e >= E) return;

    const long long s = ei[e];                  // src
    const long long d = ei[(long long)E + e];   // dst
    const float norm = dinv[s] * dinv[d];

    const float4* __restrict__ hs = (const float4*)(h + s * (long long)D_OUT);
    float* __restrict__ od = out + d * (long long)D_OUT;

#pragma unroll
    for (int it = 0; it < 2; ++it) {
        const int c4 = lane + it * 32;          // float4 index 0..63
        float4 v = hs[c4];
        atomicAdd(&od[c4 * 4 + 0], norm * v.x);
        atomicAdd(&od[c4 * 4 + 1], norm * v.y);
        atomicAdd(&od[c4 * 4 + 2], norm * v.z);
        atomicAdd(&od[c4 * 4 + 3], norm * v.w);
    }
}

// ---------------------------------------------------------------------------
// 7) finalize: out = leaky_relu(out + b, 0.2)
// ---------------------------------------------------------------------------
__global__ void finalize_kernel(float* __restrict__ out,
                                const float* __restrict__ b, long long total) {
    long long t = (long long)blockIdx.x * blockDim.x + threadIdx.x;
    if (t < total) {
        float v = out[t] + b[t & (D_OUT - 1)];
        out[t] = (v > 0.0f) ? v : 0.2f * v;
    }
}

extern "C" void kernel_launch(void* const* d_in, const int* in_sizes, int n_in,
                              void* d_out, int out_size, void* d_ws, size_t ws_size,
                              hipStream_t stream) {
    const float*     x  = (const float*)d_in[0];
    const long long* ei = (const long long*)d_in[1];  // int64 edge_index [2, E]
    const float*     W  = (const float*)d_in[2];
    const float*     b  = (const float*)d_in[3];
    float*           out = (float*)d_out;

    const int N = in_sizes[0] / D_IN;          // 100000
    const int E = in_sizes[1] / 2;             // 3200000

    // workspace: dinv [N floats], h [N*256 floats]
    float* dinv = (float*)d_ws;
    float* h    = dinv + (((size_t)N + 255) & ~(size_t)255);

    const long long total = (long long)N * D_OUT;

    // 1-3: degrees -> dinv
    deg_init_kernel<<<(N + 255) / 256, 256, 0, stream>>>(dinv, N);
    deg_count_kernel<<<(E + 255) / 256, 256, 0, stream>>>(ei, E, dinv);
    dinv_kernel<<<(N + 255) / 256, 256, 0, stream>>>(dinv, N);

    // 4: h = x @ W  (fp32 WMMA)
    dim3 ggrid((N + 127) / 128, D_OUT / 64);
    gemm_wmma_kernel<<<ggrid, 256, 0, stream>>>(x, W, h, N);

    // 5: out = dinv^2 * h  (self-loop term, also initializes d_out)
    selfloop_kernel<<<(int)((total + 255) / 256), 256, 0, stream>>>(h, dinv, out, total);

    // 6: edge aggregation
    scatter_kernel<<<(E + 7) / 8, 256, 0, stream>>>(ei, E, h, dinv, out);

    // 7: + bias, leaky relu
    finalize_kernel<<<(int)((total + 255) / 256), 256, 0, stream>>>(out, b, total);
}